// SpatialGCNLayer_51333449121796
// MI455X (gfx1250) — compile-verified
//
#include <hip/hip_runtime.h>
#include <math.h>

#define N_NODES 100000
#define N_EDGES 3200000
#define D       128
#define LN_EPS  1e-5f

typedef __attribute__((ext_vector_type(2))) float v2f;
typedef __attribute__((ext_vector_type(8))) float v8f;

// ---------------- degree / normalization ----------------

__global__ __launch_bounds__(256) void deg_init_kernel(float* __restrict__ deg) {
    int i = blockIdx.x * 256 + threadIdx.x;
    if (i < N_NODES) deg[i] = 1.0f;   // self-loop weight 1 contributes to deg
}

__global__ __launch_bounds__(256) void deg_edge_kernel(const int* __restrict__ col,
                                                       const float* __restrict__ w,
                                                       float* __restrict__ deg) {
    int e = blockIdx.x * 256 + threadIdx.x;
    if (e < N_EDGES) atomicAdd(&deg[col[e]], w[e]);
}

__global__ __launch_bounds__(256) void deg_rsqrt_kernel(float* __restrict__ deg) {
    int i = blockIdx.x * 256 + threadIdx.x;
    if (i < N_NODES) {
        float d = deg[i];
        deg[i] = (d > 0.0f) ? rsqrtf(fmaxf(d, 1e-30f)) : 0.0f;
    }
}

// ---------------- h = x @ W via fp32 WMMA ----------------
// block: 256 threads = 8 waves. Block owns 64 M-rows; wave w owns N-tile [16w,16w+16).
// x tile staged in LDS (padded stride 132 -> conflict-free column access).

#define TM 64
#define XS_STRIDE (D + 4)

__global__ __launch_bounds__(256) void gemm_wmma_kernel(const float* __restrict__ x,
                                                        const float* __restrict__ Wm,
                                                        float* __restrict__ h) {
    __shared__ float xs[TM][XS_STRIDE];

    const int t    = threadIdx.x;
    const int mblk = blockIdx.x * TM;

    // cooperative load of x[mblk .. mblk+63][0..127] : 2048 float4, 8 per thread
    #pragma unroll
    for (int i = 0; i < (TM * D / 4) / 256; ++i) {
        int f    = t + i * 256;
        int r    = f >> 5;            // 32 float4 per row
        int cb   = (f & 31) * 4;
        int grow = mblk + r;
        float4 v = make_float4(0.f, 0.f, 0.f, 0.f);
        if (grow < N_NODES) v = *(const float4*)(x + (size_t)grow * D + cb);
        *(float4*)&xs[r][cb] = v;
    }
    __syncthreads();

    const int wave = t >> 5;
    const int lane = t & 31;
    const int half = lane >> 4;       // 0: K even pair, 1: K odd pair (per ISA A/B layout)
    const int l16  = lane & 15;
    const int n0   = wave * 16;

    v8f acc[4];
    #pragma unroll
    for (int m = 0; m < 4; ++m) acc[m] = (v8f){};

    #pragma unroll
    for (int s = 0; s < 32; ++s) {
        const int k = 4 * s + 2 * half;
        v2f bv;
        bv.x = Wm[(size_t)k * D + n0 + l16];
        bv.y = Wm[(size_t)(k + 1) * D + n0 + l16];
        #pragma unroll
        for (int m = 0; m < 4; ++m) {
            const int mrow = m * 16 + l16;
            v2f av;
            av.x = xs[mrow][k];
            av.y = xs[mrow][k + 1];
            acc[m] = __builtin_amdgcn_wmma_f32_16x16x4_f32(
                false, av, false, bv, (short)0, acc[m], false, false);
        }
    }

    // D-matrix layout: VGPR v holds M = v + 8*half, N = l16
    #pragma unroll
    for (int m = 0; m < 4; ++m) {
        #pragma unroll
        for (int v = 0; v < 8; ++v) {
            int grow = mblk + m * 16 + v + 8 * half;
            if (grow < N_NODES) h[(size_t)grow * D + n0 + l16] = acc[m][v];
        }
    }
}

// ---------------- agg init: self-loop + bias ----------------

__global__ __launch_bounds__(256) void agg_init_kernel(const float* __restrict__ h,
                                                       const float* __restrict__ dis,
                                                       const float* __restrict__ b,
                                                       float* __restrict__ out) {
    int gid = blockIdx.x * 256 + threadIdx.x;          // N_NODES * 32 lanes of float4
    if (gid >= N_NODES * 32) return;
    int node = gid >> 5;
    int c4   = (gid & 31) * 4;
    float s  = dis[node];
    s = s * s;                                         // dis[i]*1*dis[i]
    float4 hv = *(const float4*)(h + (size_t)node * D + c4);
    float4 bv = *(const float4*)(b + c4);
    float4 o;
    o.x = s * hv.x + bv.x;
    o.y = s * hv.y + bv.y;
    o.z = s * hv.z + bv.z;
    o.w = s * hv.w + bv.w;
    *(float4*)(out + (size_t)node * D + c4) = o;
}

// ---------------- edge scatter: one wave per edge ----------------

__global__ __launch_bounds__(256) void edge_scatter_kernel(const int* __restrict__ ridx,
                                                           const int* __restrict__ cidx,
                                                           const float* __restrict__ w,
                                                           const float* __restrict__ dis,
                                                           const float* __restrict__ h,
                                                           float* __restrict__ out) {
    int e = blockIdx.x * 8 + (threadIdx.x >> 5);
    if (e >= N_EDGES) return;
    int lane = threadIdx.x & 31;
    int r = ridx[e];
    int c = cidx[e];
    float coef = dis[r] * w[e] * dis[c];
    float4 hv = *(const float4*)(h + (size_t)r * D + lane * 4);
    float* o = out + (size_t)c * D + lane * 4;
    atomicAdd(o + 0, coef * hv.x);
    atomicAdd(o + 1, coef * hv.y);
    atomicAdd(o + 2, coef * hv.z);
    atomicAdd(o + 3, coef * hv.w);
}

// ---------------- LayerNorm + ReLU, in place, one wave per node ----------------

__global__ __launch_bounds__(256) void ln_relu_kernel(const float* __restrict__ gamma,
                                                      const float* __restrict__ beta,
                                                      float* __restrict__ out) {
    int node = blockIdx.x * 8 + (threadIdx.x >> 5);
    if (node >= N_NODES) return;
    int lane = threadIdx.x & 31;
    float4 v = *(const float4*)(out + (size_t)node * D + lane * 4);
    float s  = v.x + v.y + v.z + v.w;
    float s2 = v.x * v.x + v.y * v.y + v.z * v.z + v.w * v.w;
    #pragma unroll
    for (int m = 16; m >= 1; m >>= 1) {
        s  += __shfl_xor(s,  m, 32);
        s2 += __shfl_xor(s2, m, 32);
    }
    float mu  = s * (1.0f / D);
    float var = s2 * (1.0f / D) - mu * mu;
    float rs  = rsqrtf(var + LN_EPS);
    float4 g  = *(const float4*)(gamma + lane * 4);
    float4 be = *(const float4*)(beta + lane * 4);
    float4 o;
    o.x = fmaxf((v.x - mu) * rs * g.x + be.x, 0.0f);
    o.y = fmaxf((v.y - mu) * rs * g.y + be.y, 0.0f);
    o.z = fmaxf((v.z - mu) * rs * g.z + be.z, 0.0f);
    o.w = fmaxf((v.w - mu) * rs * g.w + be.w, 0.0f);
    *(float4*)(out + (size_t)node * D + lane * 4) = o;
}

// ---------------- launcher ----------------

extern "C" void kernel_launch(void* const* d_in, const int* in_sizes, int n_in,
                              void* d_out, int out_size, void* d_ws, size_t ws_size,
                              hipStream_t stream) {
    const float* x      = (const float*)d_in[0];
    const int*   eidx   = (const int*)d_in[1];       // [2, E] flat
    const float* ew     = (const float*)d_in[2];
    const float* Wm     = (const float*)d_in[3];
    const float* b      = (const float*)d_in[4];
    const float* gamma  = (const float*)d_in[5];
    const float* beta   = (const float*)d_in[6];
    float*       out    = (float*)d_out;

    const int* row = eidx;
    const int* col = eidx + N_EDGES;

    float* h   = (float*)d_ws;                 // N_NODES * D floats
    float* deg = h + (size_t)N_NODES * D;      // N_NODES floats (becomes dis in place)

    // 1) degree with self-loops
    deg_init_kernel<<<(N_NODES + 255) / 256, 256, 0, stream>>>(deg);
    deg_edge_kernel<<<(N_EDGES + 255) / 256, 256, 0, stream>>>(col, ew, deg);
    deg_rsqrt_kernel<<<(N_NODES + 255) / 256, 256, 0, stream>>>(deg);

    // 2) h = x @ W  (fp32 WMMA)
    gemm_wmma_kernel<<<(N_NODES + TM - 1) / TM, 256, 0, stream>>>(x, Wm, h);

    // 3) agg = dis^2 * h (self-loop) + b, built in d_out
    agg_init_kernel<<<(N_NODES * 32 + 255) / 256, 256, 0, stream>>>(h, deg, b, out);

    // 4) agg += dis[r]*w*dis[c] * h[r]  (wave per edge, float atomics)
    edge_scatter_kernel<<<(N_EDGES + 7) / 8, 256, 0, stream>>>(row, col, ew, deg, h, out);

    // 5) LayerNorm + ReLU in place
    ln_relu_kernel<<<(N_NODES + 7) / 8, 256, 0, stream>>>(gamma, beta, out);
}